// WeightQuantizer_23158463660206
// MI455X (gfx1250) — compile-verified
//
#include <hip/hip_runtime.h>

typedef float v2f __attribute__((ext_vector_type(2)));
typedef float v8f __attribute__((ext_vector_type(8)));

#define GROUP 128
#define TILES_PER_WAVE 8   // 8 tiles = 16 groups = 8KB input per wave

// ds_swizzle group-of-32 xor patterns: offset = (xor<<10)|(or<<5)|and, and=0x1f
#define SWZ_X1 0x041F
#define SWZ_X2 0x081F
#define SWZ_X4 0x101F
#define SWZ_X8 0x201F

__device__ __forceinline__ float maxswz(float x, float y) {
    return fmaxf(x, y);
}

// Each 16x16 A-tile covers 2 groups (rows 0-7 = g0's 8x16 view, rows 8-15 =
// g1's). Step 1: Z = X * (H16 * 128^-1/2) via 4x v_wmma_f32_16x16x4_f32.
// Step 2: Y = H8 * Z as an in-register butterfly across the 8 accumulator
// components (C layout puts a group's 8 sub-rows in one lane's 8 VGPRs).
// Then per-group absmax via ds_swizzle xor-swaps within each 16-lane half.
__global__ __launch_bounds__(256) void fwht_quant_kernel(
    const float* __restrict__ w,
    float* __restrict__ scale_out,
    float* __restrict__ zp_out,
    float* __restrict__ wq_out,
    int ngroups)
{
    const int lane = threadIdx.x & 31;
    const int wave = threadIdx.x >> 5;
    const int waveId   = blockIdx.x * 8 + wave;
    const int tileBase = waveId * TILES_PER_WAVE;

    const int n  = lane & 15;   // N column / tile row index
    const int hi = lane >> 4;   // K-half selector (A/B), group selector (C)

    // B = H16 * 1/sqrt(128), built once per wave. Sign folded into the FP32
    // sign bit: H16[k][n] = (-1)^popc(k&n). Layout mirrors the 32-bit A
    // table: lanes 0-15 hold K=4s+{0,1}, lanes 16-31 hold K=4s+{2,3}.
    const unsigned cbits = 0x3DB504F3u;       // 128^-0.5 as f32
    v2f B[4];
#pragma unroll
    for (int s = 0; s < 4; ++s) {
        const unsigned k0 = 4u * s + 2u * hi;
        B[s].x = __uint_as_float(cbits ^ ((unsigned)__popc(k0 & n) << 31));
        B[s].y = __uint_as_float(cbits ^ ((unsigned)__popc((k0 + 1u) & n) << 31));
    }

#pragma unroll 1
    for (int t = 0; t < TILES_PER_WAVE; ++t) {
        const int g0 = (tileBase + t) * 2;    // fits 32-bit (<= 352256)
        if (g0 >= ngroups) break;             // wave-uniform exit
        const int g1valid = (g0 + 1) < ngroups;
        const int g1 = g1valid ? (g0 + 1) : g0;  // clamp keeps EXEC full

        // A row r = n; rows 0-7 -> g0, rows 8-15 -> g1. All element offsets
        // fit in 32 bits (max 45M floats = 180MB).
        const int rowg = (n < 8) ? g0 : g1;
        const float* rowp = w + (unsigned)(rowg * GROUP + (n & 7) * 16 + 2 * hi);

        // Prefetch next tile's rows into GL2 (speculative; OOB dropped).
        __builtin_prefetch(rowp + 2 * GROUP, 0, 0);

        v8f acc = {};
#pragma unroll
        for (int s = 0; s < 4; ++s) {
            v2f a = __builtin_nontemporal_load((const v2f*)(rowp + 4 * s));
            acc = __builtin_amdgcn_wmma_f32_16x16x4_f32(
                false, a, false, B[s], (short)0, acc, false, false);
        }

        // H8 butterfly across accumulator components (Sylvester order).
        float z0 = acc[0], z1 = acc[1], z2 = acc[2], z3 = acc[3];
        float z4 = acc[4], z5 = acc[5], z6 = acc[6], z7 = acc[7];
        float t0 = z0 + z1, t1 = z0 - z1, t2 = z2 + z3, t3 = z2 - z3;
        float t4 = z4 + z5, t5 = z4 - z5, t6 = z6 + z7, t7 = z6 - z7;
        float u0 = t0 + t2, u2 = t0 - t2, u1 = t1 + t3, u3 = t1 - t3;
        float u4 = t4 + t6, u6 = t4 - t6, u5 = t5 + t7, u7 = t5 - t7;
        float y[8];
        y[0] = u0 + u4; y[4] = u0 - u4; y[1] = u1 + u5; y[5] = u1 - u5;
        y[2] = u2 + u6; y[6] = u2 - u6; y[3] = u3 + u7; y[7] = u3 - u7;

        // Per-group absmax: 8 in-lane, then xor-swap reduce within each
        // 16-lane half via ds_swizzle (immediate patterns, no index math).
        float m = fabsf(y[0]);
#pragma unroll
        for (int j = 1; j < 8; ++j) m = fmaxf(m, fabsf(y[j]));
        m = maxswz(m, __uint_as_float(
                __builtin_amdgcn_ds_swizzle(__float_as_uint(m), SWZ_X1)));
        m = maxswz(m, __uint_as_float(
                __builtin_amdgcn_ds_swizzle(__float_as_uint(m), SWZ_X2)));
        m = maxswz(m, __uint_as_float(
                __builtin_amdgcn_ds_swizzle(__float_as_uint(m), SWZ_X4)));
        m = maxswz(m, __uint_as_float(
                __builtin_amdgcn_ds_swizzle(__float_as_uint(m), SWZ_X8)));

        const float scale = fmaxf(m / 7.0f, 1e-8f);

        const int g = hi ? g1 : g0;
        if ((hi == 0) || g1valid) {
            float* wg = wq_out + (unsigned)(g * GROUP + n);
#pragma unroll
            for (int j = 0; j < 8; ++j) {
                float q = rintf(y[j] / scale);
                q = fminf(fmaxf(q, -8.0f), 7.0f);
                __builtin_nontemporal_store(q, wg + j * 16);
            }
            if (n == 0) {
                scale_out[g] = scale;
                zp_out[g]    = 0.0f;
            }
        }
    }
}

extern "C" void kernel_launch(void* const* d_in, const int* in_sizes, int n_in,
                              void* d_out, int out_size, void* d_ws, size_t ws_size,
                              hipStream_t stream) {
    (void)n_in; (void)d_ws; (void)ws_size; (void)out_size;
    const float* w = (const float*)d_in[0];
    const int total   = in_sizes[0];        // 4096*11008
    const int ngroups = total / GROUP;      // 352256

    float* scale_out = (float*)d_out;
    float* zp_out    = scale_out + ngroups;
    float* wq_out    = zp_out + ngroups;

    const int ntiles = (ngroups + 1) / 2;                       // 176128
    const int waves  = (ntiles + TILES_PER_WAVE - 1) / TILES_PER_WAVE;
    const int blocks = (waves + 7) / 8;     // 8 waves (256 threads) per block
    fwht_quant_kernel<<<blocks, 256, 0, stream>>>(w, scale_out, zp_out, wq_out,
                                                  ngroups);
}